// GAT_v2_Res_Model_52690658787912
// MI455X (gfx1250) — compile-verified
//
#include <hip/hip_runtime.h>
#include <cmath>

#define N_NODES 100000
#define N_EDGES 1600000
#define EPSV 1e-16f

typedef float v2f __attribute__((ext_vector_type(2)));
typedef float v8f __attribute__((ext_vector_type(8)));

__device__ __forceinline__ float leaky(float x) { return x > 0.f ? x : 0.2f * x; }

// float atomic-max via integer atomics (works for all sign combos; init = -inf)
__device__ __forceinline__ void atomicMaxF(float* addr, float val) {
  if (val >= 0.f) atomicMax((int*)addr, __float_as_int(val));
  else            atomicMin((unsigned int*)addr, __float_as_uint(val));
}

// ---------------- GEMM: XL[N,M] = X[N,K] @ W[K,M], fp32 WMMA 16x16x4 ----------------
template<int K, int M>
__global__ __launch_bounds__(256)
void gemm_wmma_f32(const float* __restrict__ X, const float* __restrict__ W,
                   float* __restrict__ XL)
{
  constexpr int KP = K + 4;            // LDS padding -> conflict-free column access
  __shared__ float As[16 * KP];
  const int row0 = blockIdx.x * 16;

  // cooperative, coalesced load of the 16 x K A-tile
  for (int i = threadIdx.x; i < (16 * K) / 4; i += blockDim.x) {
    const int flat = i * 4;
    const int r = flat / K, c = flat % K;
    const float4 v = *(const float4*)(X + (size_t)(row0 + r) * K + c);
    *(float4*)(As + r * KP + c) = v;
  }
  __syncthreads();

  const int wave = threadIdx.x >> 5;
  const int lane = threadIdx.x & 31;
  const int m    = lane & 15;   // A row / B col / C col
  const int hi   = lane >> 4;   // half-wave selects K pair (A/B) and row block (C)

  for (int ct = wave; ct < M / 16; ct += 8) {
    const int col0 = ct * 16;
    v8f c = {0.f, 0.f, 0.f, 0.f, 0.f, 0.f, 0.f, 0.f};
    #pragma unroll 8
    for (int k0 = 0; k0 < K; k0 += 4) {
      const int ka = k0 + hi * 2;
      // A 16x4: lane m holds A[m][ka], A[m][ka+1]
      v2f a = *(const v2f*)(As + m * KP + ka);
      // B 4x16: lane m holds B[ka][col0+m], B[ka+1][col0+m]
      v2f b;
      b.x = W[(size_t)ka * M + col0 + m];
      b.y = W[(size_t)(ka + 1) * M + col0 + m];
      c = __builtin_amdgcn_wmma_f32_16x16x4_f32(false, a, false, b, (short)0, c, false, false);
    }
    // C 16x16 f32: VGPR i -> row (hi*8 + i), lane -> column
    const int rbase = row0 + hi * 8;
    #pragma unroll
    for (int i = 0; i < 8; ++i)
      XL[(size_t)(rbase + i) * M + col0 + m] = c[i];
  }
}

// ---------------- per-edge attention scores + segment max ----------------
// CPL = channels per lane (M = 32*CPL, head width C = 8*CPL, head = lane/8)
template<int CPL>
__global__ __launch_bounds__(256)
void edge_scores(const int* __restrict__ src, const int* __restrict__ dst,
                 const float* __restrict__ xl, const float* __restrict__ att,
                 float* __restrict__ alpha, float* __restrict__ amax)
{
  const int e    = (int)((blockIdx.x * (size_t)blockDim.x + threadIdx.x) >> 5);
  const int lane = threadIdx.x & 31;
  if (e >= N_EDGES) return;
  const int s = src[e], d = dst[e];
  constexpr int M = 32 * CPL;
  const float* xi = xl + (size_t)d * M;
  const float* xj = xl + (size_t)s * M;
  float p = 0.f;
  #pragma unroll
  for (int v = 0; v < CPL / 4; ++v) {
    const int f = lane * CPL + v * 4;
    const float4 a4 = *(const float4*)(att + f);
    const float4 i4 = *(const float4*)(xi + f);
    const float4 j4 = *(const float4*)(xj + f);
    p += leaky(i4.x + j4.x) * a4.x + leaky(i4.y + j4.y) * a4.y
       + leaky(i4.z + j4.z) * a4.z + leaky(i4.w + j4.w) * a4.w;
  }
  // reduce the 8 lanes of each head (wave32)
  p += __shfl_xor(p, 1, 32);
  p += __shfl_xor(p, 2, 32);
  p += __shfl_xor(p, 4, 32);
  if ((lane & 7) == 0) {
    const int h = lane >> 3;
    alpha[(size_t)e * 4 + h] = p;
    atomicMaxF(amax + (size_t)d * 4 + h, p);
  }
}

// ---------------- exp(alpha - max) + segment sum of exps ----------------
__global__ __launch_bounds__(256)
void edge_exp(const int* __restrict__ dst, const float* __restrict__ amax,
              float* __restrict__ alpha_ex, float* __restrict__ denom)
{
  const size_t t = blockIdx.x * (size_t)blockDim.x + threadIdx.x;
  if (t >= (size_t)N_EDGES * 4) return;
  const int e = (int)(t >> 2), h = (int)(t & 3);
  const int d = dst[e];
  float am = amax[(size_t)d * 4 + h];
  if (isinf(am)) am = 0.f;
  const float ex = __expf(alpha_ex[t] - am);
  alpha_ex[t] = ex;
  unsafeAtomicAdd(denom + (size_t)d * 4 + h, ex);
}

// ---------------- scatter: acc[dst] += xl[src] * ex ----------------
template<int CPL>
__global__ __launch_bounds__(256)
void edge_aggregate(const int* __restrict__ src, const int* __restrict__ dst,
                    const float* __restrict__ xl, const float* __restrict__ ex,
                    float* __restrict__ acc)
{
  const int e    = (int)((blockIdx.x * (size_t)blockDim.x + threadIdx.x) >> 5);
  const int lane = threadIdx.x & 31;
  if (e >= N_EDGES) return;
  const int s = src[e], d = dst[e];
  constexpr int M = 32 * CPL;
  const float w = ex[(size_t)e * 4 + (lane >> 3)];
  const float* xj = xl + (size_t)s * M;
  float* ad = acc + (size_t)d * M;
  #pragma unroll
  for (int v = 0; v < CPL / 4; ++v) {
    const int f = lane * CPL + v * 4;
    const float4 j4 = *(const float4*)(xj + f);
    unsafeAtomicAdd(ad + f + 0, j4.x * w);
    unsafeAtomicAdd(ad + f + 1, j4.y * w);
    unsafeAtomicAdd(ad + f + 2, j4.z * w);
    unsafeAtomicAdd(ad + f + 3, j4.w * w);
  }
}

// ---------------- epilogues ----------------
// layers 0/1: h = elu( 0.9*(acc/denom' + b) + 0.1*x0 ), width 128
__global__ __launch_bounds__(256)
void epilogue01(const float* __restrict__ acc, const float* __restrict__ denom,
                const float* __restrict__ b, const float* __restrict__ x0,
                float* __restrict__ h)
{
  const size_t t = blockIdx.x * (size_t)blockDim.x + threadIdx.x;
  if (t >= (size_t)N_NODES * 128) return;
  const int n = (int)(t >> 7);
  const int f = (int)(t & 127);
  const int hd = f >> 5;
  float v = acc[t] / (denom[(size_t)n * 4 + hd] + EPSV) + b[f];
  v = v * 0.9f + x0[t] * 0.1f;
  h[t] = v > 0.f ? v : (__expf(v) - 1.f);
}

// layer 2: out = mean_heads(acc/denom') + b2, width 64
__global__ __launch_bounds__(256)
void epilogue2(const float* __restrict__ acc, const float* __restrict__ denom,
               const float* __restrict__ b, float* __restrict__ out)
{
  const size_t t = blockIdx.x * (size_t)blockDim.x + threadIdx.x;
  if (t >= (size_t)N_NODES * 64) return;
  const int n = (int)(t >> 6);
  const int c = (int)(t & 63);
  float s = 0.f;
  #pragma unroll
  for (int h = 0; h < 4; ++h)
    s += acc[(size_t)n * 256 + h * 64 + c] / (denom[(size_t)n * 4 + h] + EPSV);
  out[t] = s * 0.25f + b[c];
}

__global__ __launch_bounds__(256)
void fill_f32(float* __restrict__ p, float v, long n)
{
  const long i = blockIdx.x * (long)blockDim.x + threadIdx.x;
  if (i < n) p[i] = v;
}

extern "C" void kernel_launch(void* const* d_in, const int* in_sizes, int n_in,
                              void* d_out, int out_size, void* d_ws, size_t ws_size,
                              hipStream_t stream)
{
  (void)in_sizes; (void)n_in; (void)out_size; (void)ws_size;
  const float* x    = (const float*)d_in[0];
  const int*   ei   = (const int*)d_in[1];   // [2,E]; src row then dst row
  const float* W0   = (const float*)d_in[2];
  const float* b0   = (const float*)d_in[3];
  const float* att0 = (const float*)d_in[4];
  const float* W1   = (const float*)d_in[5];
  const float* b1   = (const float*)d_in[6];
  const float* att1 = (const float*)d_in[7];
  const float* W2   = (const float*)d_in[8];
  const float* b2   = (const float*)d_in[9];
  const float* att2 = (const float*)d_in[10];
  const int* src = ei;
  const int* dst = ei + N_EDGES;

  float* ws    = (float*)d_ws;
  float* xl0   = ws;                               // N*128 (layer-0 features == residual x0)
  float* hbuf  = xl0   + (size_t)N_NODES * 128;    // N*128 (elu outputs feeding next layer)
  float* xl    = hbuf  + (size_t)N_NODES * 128;    // N*256 (layer 1/2 transformed features)
  float* acc   = xl    + (size_t)N_NODES * 256;    // N*256 (scatter accumulator)
  float* alpha = acc   + (size_t)N_NODES * 256;    // E*4   (scores, then exps)
  float* amax  = alpha + (size_t)N_EDGES * 4;      // N*4
  float* denom = amax  + (size_t)N_NODES * 4;      // N*4

  const int TB     = 256;
  const int gGemm  = N_NODES / 16;                 // 6250
  const int gEdgeW = (N_EDGES * 32) / TB;          // 200000 (1 wave / edge)
  const int gE4    = (N_EDGES * 4) / TB;           // 25000
  const int gN128  = (N_NODES * 128) / TB;         // 50000
  const int gN64   = (N_NODES * 64) / TB;          // 25000
  const int gN4    = (N_NODES * 4 + TB - 1) / TB;

  // ---------------- layer 0 (K=256 -> M=128) ----------------
  gemm_wmma_f32<256,128><<<gGemm, TB, 0, stream>>>(x, W0, xl0);
  fill_f32<<<gN4, TB, 0, stream>>>(amax, -INFINITY, (long)N_NODES * 4);
  hipMemsetAsync(denom, 0, (size_t)N_NODES * 4 * sizeof(float), stream);
  hipMemsetAsync(acc,   0, (size_t)N_NODES * 128 * sizeof(float), stream);
  edge_scores<4><<<gEdgeW, TB, 0, stream>>>(src, dst, xl0, att0, alpha, amax);
  edge_exp<<<gE4, TB, 0, stream>>>(dst, amax, alpha, denom);
  edge_aggregate<4><<<gEdgeW, TB, 0, stream>>>(src, dst, xl0, alpha, acc);
  epilogue01<<<gN128, TB, 0, stream>>>(acc, denom, b0, xl0, hbuf);

  // ---------------- layer 1 (K=128 -> M=128) ----------------
  gemm_wmma_f32<128,128><<<gGemm, TB, 0, stream>>>(hbuf, W1, xl);
  fill_f32<<<gN4, TB, 0, stream>>>(amax, -INFINITY, (long)N_NODES * 4);
  hipMemsetAsync(denom, 0, (size_t)N_NODES * 4 * sizeof(float), stream);
  hipMemsetAsync(acc,   0, (size_t)N_NODES * 128 * sizeof(float), stream);
  edge_scores<4><<<gEdgeW, TB, 0, stream>>>(src, dst, xl, att1, alpha, amax);
  edge_exp<<<gE4, TB, 0, stream>>>(dst, amax, alpha, denom);
  edge_aggregate<4><<<gEdgeW, TB, 0, stream>>>(src, dst, xl, alpha, acc);
  epilogue01<<<gN128, TB, 0, stream>>>(acc, denom, b1, xl0, hbuf);

  // ---------------- layer 2 (K=128 -> M=256, mean over heads) ----------------
  gemm_wmma_f32<128,256><<<gGemm, TB, 0, stream>>>(hbuf, W2, xl);
  fill_f32<<<gN4, TB, 0, stream>>>(amax, -INFINITY, (long)N_NODES * 4);
  hipMemsetAsync(denom, 0, (size_t)N_NODES * 4 * sizeof(float), stream);
  hipMemsetAsync(acc,   0, (size_t)N_NODES * 256 * sizeof(float), stream);
  edge_scores<8><<<gEdgeW, TB, 0, stream>>>(src, dst, xl, att2, alpha, amax);
  edge_exp<<<gE4, TB, 0, stream>>>(dst, amax, alpha, denom);
  edge_aggregate<8><<<gEdgeW, TB, 0, stream>>>(src, dst, xl, alpha, acc);
  epilogue2<<<gN64, TB, 0, stream>>>(acc, denom, b2, (float*)d_out);
}